// GraphModule_56642028699712
// MI455X (gfx1250) — compile-verified
//
#include <hip/hip_runtime.h>
#include <hip/hip_bf16.h>

typedef __attribute__((ext_vector_type(2))) float v2f;
typedef __attribute__((ext_vector_type(8))) float v8f;

#define D_IN    128
#define D_HID   256

// ---------------------------------------------------------------------------
// Degree / normalization kernels
// ---------------------------------------------------------------------------
__global__ void deg_init_kernel(float* __restrict__ deg, int n) {
    int i = blockIdx.x * blockDim.x + threadIdx.x;
    if (i < n) deg[i] = 1.0f;  // self-loop weight 1
}

__global__ void deg_scatter_kernel(const int* __restrict__ dst,
                                   const float* __restrict__ ew,
                                   float* __restrict__ deg, int E) {
    int e = blockIdx.x * blockDim.x + threadIdx.x;
    if (e < E) atomicAdd(&deg[dst[e]], ew[e]);
}

__global__ void dinv_kernel(const float* __restrict__ deg,
                            float* __restrict__ dinv, int n) {
    int i = blockIdx.x * blockDim.x + threadIdx.x;
    if (i < n) {
        float d = deg[i];
        dinv[i] = (d > 0.0f) ? rsqrtf(fmaxf(d, 1e-12f)) : 0.0f;
    }
}

// ---------------------------------------------------------------------------
// Pack W[K,N] row-major -> pair-interleaved Wp[(K/2)*N] of float2:
//   Wp[kp*N + n] = { W[2kp][n], W[2kp+1][n] }
// This makes the per-lane WMMA B-fragment (two consecutive K values of one
// column) a single aligned 64-bit load, and the LDS slab copy a straight
// byte copy (async-LDS friendly).
// ---------------------------------------------------------------------------
template <int N, int K>
__global__ void pack_w_kernel(const float* __restrict__ W, v2f* __restrict__ Wp) {
    int j = blockIdx.x * blockDim.x + threadIdx.x;
    if (j < (K / 2) * N) {
        int kp = j / N, n = j % N;
        v2f v;
        v.x = W[(2 * kp) * N + n];
        v.y = W[(2 * kp + 1) * N + n];
        Wp[j] = v;
    }
}

// ---------------------------------------------------------------------------
// WMMA fp32 GEMM: C[M,N] = A[M,K] * W[K,N]   (W given pair-packed as Wp)
// One wave per 16-row M block computing the full N strip (NT = N/16 tiles).
// Wp is streamed through two 32KB LDS slabs, double-buffered with
// GLOBAL_LOAD_ASYNC_TO_LDS_B128 (ASYNCcnt) so slab s+1 loads behind slab s's
// WMMA work. A-fragment (16x4 f32): lanes 0-15 hold K=k0,k0+1; lanes 16-31
// hold K=k0+2,k0+3. C/D: VGPR r -> row r / 8+r per lane half, col = lane&15.
// ---------------------------------------------------------------------------
template <int N, int K>
__global__ __launch_bounds__(128) void gemm_wmma_f32_kernel(
    const float* __restrict__ A, const v2f* __restrict__ Wp,
    float* __restrict__ C, int M) {
    constexpr int KS = 8192 / N;              // K rows per slab -> 32KB slab
    constexpr int NT = N / 16;
    constexpr int NS = K / KS;                // 4 slabs
    constexpr int SLAB_V2 = (KS / 2) * N;     // v2f per slab
    constexpr int SLAB_BYTES = KS * N * 4;    // 32768
    __shared__ v2f sW[2][SLAB_V2];

    const int wave  = threadIdx.x >> 5;
    const int lane  = threadIdx.x & 31;
    const int m0    = (blockIdx.x * 4 + wave) * 16;
    const bool active = (m0 < M);             // wave-uniform -> EXEC all-1s
    const int row   = m0 + (lane & 15);
    const int khalf = (lane >> 4) * 2;        // 0 / 2 : K offset of A frag
    const int khl   = lane >> 4;              // 0 / 1 : pair index offset
    const int nlane = lane & 15;

    const unsigned lds0 = (unsigned)(uintptr_t)&sW[0][0];
    const unsigned lds1 = (unsigned)(uintptr_t)&sW[1][0];
    const char* wbytes = (const char*)Wp;

    auto issue_slab = [&](int s) {
        unsigned dst = ((s & 1) ? lds1 : lds0) + (unsigned)threadIdx.x * 16u;
        unsigned src = (unsigned)s * (unsigned)SLAB_BYTES + (unsigned)threadIdx.x * 16u;
#pragma unroll
        for (int i = 0; i < SLAB_BYTES / (128 * 16); ++i) {
            asm volatile("global_load_async_to_lds_b128 %0, %1, %2 offset:0"
                         :: "v"(dst), "v"(src), "s"(wbytes) : "memory");
            dst += 128u * 16u;
            src += 128u * 16u;
        }
    };

    v8f acc[NT];
#pragma unroll
    for (int t = 0; t < NT; ++t)
#pragma unroll
        for (int j = 0; j < 8; ++j) acc[t][j] = 0.0f;

    const float* Arow = A + (size_t)(active ? row : 0) * K;

    issue_slab(0);
    asm volatile("s_wait_asynccnt 0" ::: "memory");
    __syncthreads();

    for (int s = 0; s < NS; ++s) {
        if (s + 1 < NS) issue_slab(s + 1);    // prefetch next slab (other buffer)
        const v2f* wbuf = sW[s & 1];
        if (active) {
#pragma unroll 2
            for (int kk = 0; kk < KS; kk += 4) {
                v2f a = *(const v2f*)(Arow + s * KS + kk + khalf);
                const v2f* wp = wbuf + (kk / 2 + khl) * N + nlane;
#pragma unroll
                for (int t = 0; t < NT; ++t) {
                    v2f b = wp[t * 16];       // single ds_load_b64
                    acc[t] = __builtin_amdgcn_wmma_f32_16x16x4_f32(
                        false, a, false, b, (short)0, acc[t], false, false);
                }
            }
        }
        asm volatile("s_wait_asynccnt 0" ::: "memory");
        __syncthreads();
    }

    if (!active) return;
    const int rbase = m0 + ((lane >> 4) * 8);
#pragma unroll
    for (int t = 0; t < NT; ++t)
#pragma unroll
        for (int r = 0; r < 8; ++r)
            C[(size_t)(rbase + r) * N + t * 16 + nlane] = acc[t][r];
}

// ---------------------------------------------------------------------------
// Aggregation: agg[i,:] = dinv[i]^2 * h[i,:]  (self-loop term, also init)
// ---------------------------------------------------------------------------
template <int F>
__global__ void agg_init_self_kernel(const float* __restrict__ h,
                                     const float* __restrict__ dinv,
                                     float* __restrict__ agg, int total) {
    int i = blockIdx.x * blockDim.x + threadIdx.x;
    if (i < total) {
        float dv = dinv[i / F];
        agg[i] = dv * dv * h[i];
    }
}

// agg[dst,:] += dinv[src]*ew*dinv[dst] * h[src,:]   (float4 gather, atomic scatter)
template <int F>
__global__ __launch_bounds__(256) void edge_scatter_kernel(
    const int* __restrict__ src, const int* __restrict__ dst,
    const float* __restrict__ ew, const float* __restrict__ dinv,
    const float* __restrict__ h, float* __restrict__ agg, int E) {
    constexpr int LPE = F / 4;        // lanes per edge (float4 each)
    constexpr int EPB = 256 / LPE;    // edges per block
    const int e  = blockIdx.x * EPB + threadIdx.x / LPE;
    const int fl = (threadIdx.x % LPE) * 4;
    if (e >= E) return;
    const int s = src[e], d = dst[e];
    const float norm = dinv[s] * ew[e] * dinv[d];
    const float4 hv = *(const float4*)(h + (size_t)s * F + fl);
    float* ap = agg + (size_t)d * F + fl;
    atomicAdd(ap + 0, norm * hv.x);
    atomicAdd(ap + 1, norm * hv.y);
    atomicAdd(ap + 2, norm * hv.z);
    atomicAdd(ap + 3, norm * hv.w);
}

// in-place: a = relu(a + b[f])
template <int F>
__global__ void bias_relu_kernel(float* __restrict__ a,
                                 const float* __restrict__ b, int total) {
    int i = blockIdx.x * blockDim.x + threadIdx.x;
    if (i < total) {
        float v = a[i] + b[i & (F - 1)];
        a[i] = v > 0.0f ? v : 0.0f;
    }
}

// ---------------------------------------------------------------------------
// LayerNorm over 128 features: one wave per node, float4 per lane.
// out = (x - mu) * rsqrt(var + eps) * gamma + beta,  x = agg + b2
// ---------------------------------------------------------------------------
__global__ __launch_bounds__(256) void layernorm_kernel(
    const float* __restrict__ agg, const float* __restrict__ b2,
    const float* __restrict__ gamma, const float* __restrict__ beta,
    float* __restrict__ out, int n) {
    const int wave = threadIdx.x >> 5;
    const int lane = threadIdx.x & 31;
    const int node = blockIdx.x * 8 + wave;
    if (node >= n) return;

    const float4 v  = *(const float4*)(agg + (size_t)node * 128 + lane * 4);
    const float4 bb = *(const float4*)(b2 + lane * 4);
    float4 x;
    x.x = v.x + bb.x; x.y = v.y + bb.y; x.z = v.z + bb.z; x.w = v.w + bb.w;

    float s  = x.x + x.y + x.z + x.w;
    float s2 = x.x * x.x + x.y * x.y + x.z * x.z + x.w * x.w;
#pragma unroll
    for (int off = 16; off > 0; off >>= 1) {
        s  += __shfl_xor(s,  off, 32);
        s2 += __shfl_xor(s2, off, 32);
    }
    const float mu  = s * (1.0f / 128.0f);
    const float var = s2 * (1.0f / 128.0f) - mu * mu;
    const float inv = rsqrtf(var + 1e-5f);

    const float4 g  = *(const float4*)(gamma + lane * 4);
    const float4 bt = *(const float4*)(beta + lane * 4);
    float4 o;
    o.x = (x.x - mu) * inv * g.x + bt.x;
    o.y = (x.y - mu) * inv * g.y + bt.y;
    o.z = (x.z - mu) * inv * g.z + bt.z;
    o.w = (x.w - mu) * inv * g.w + bt.w;
    *(float4*)(out + (size_t)node * 128 + lane * 4) = o;
}

// ---------------------------------------------------------------------------
extern "C" void kernel_launch(void* const* d_in, const int* in_sizes, int n_in,
                              void* d_out, int out_size, void* d_ws, size_t ws_size,
                              hipStream_t stream) {
    const float* x     = (const float*)d_in[0];        // [N,128]
    const int*   eidx  = (const int*)d_in[1];          // [2,E]
    const float* ew    = (const float*)d_in[2];        // [E]
    const float* W1    = (const float*)d_in[3];        // [128,256]
    const float* b1    = (const float*)d_in[4];        // [256]
    const float* W2    = (const float*)d_in[5];        // [256,128]
    const float* b2    = (const float*)d_in[6];        // [128]
    const float* gamma = (const float*)d_in[7];        // [128]
    const float* beta  = (const float*)d_in[8];        // [128]

    const int N = in_sizes[0] / D_IN;                  // 50000
    const int E = in_sizes[2];                         // 800000
    const int* src = eidx;
    const int* dst = eidx + E;

    float* ws   = (float*)d_ws;
    float* deg  = ws;                                  // [N]
    float* dinv = ws + N;                              // [N]
    float* hbuf = ws + 2 * (size_t)N;                  // [N,256] (h1, then h2 in [N,128])
    float* abuf = hbuf + (size_t)N * D_HID;            // [N,256] (agg1/hrelu, then agg2)
    v2f*   wp1  = (v2f*)(abuf + (size_t)N * D_HID);    // [(128/2)*256]
    v2f*   wp2  = wp1 + (D_IN / 2) * D_HID;            // [(256/2)*128]

    const int mblocks   = (N + 15) / 16;               // 3125
    const int gemm_grid = (mblocks + 3) / 4;           // 4 waves/block

    // degree + normalization (runs concurrently with nothing; tiny)
    deg_init_kernel<<<(N + 255) / 256, 256, 0, stream>>>(deg, N);
    deg_scatter_kernel<<<(E + 255) / 256, 256, 0, stream>>>(dst, ew, deg, E);
    dinv_kernel<<<(N + 255) / 256, 256, 0, stream>>>(deg, dinv, N);

    // pack weights into pair-interleaved layout (128KB each, one-shot)
    pack_w_kernel<D_HID, D_IN><<<((D_IN / 2) * D_HID + 255) / 256, 256, 0, stream>>>(W1, wp1);
    pack_w_kernel<D_IN, D_HID><<<((D_HID / 2) * D_IN + 255) / 256, 256, 0, stream>>>(W2, wp2);

    // ---- layer 1: h1 = x @ W1 ; agg1 = scatter(norm * h1[src]) ; relu(+b1)
    gemm_wmma_f32_kernel<D_HID, D_IN><<<gemm_grid, 128, 0, stream>>>(x, wp1, hbuf, N);
    agg_init_self_kernel<D_HID><<<((size_t)N * D_HID + 255) / 256, 256, 0, stream>>>(
        hbuf, dinv, abuf, N * D_HID);
    edge_scatter_kernel<D_HID><<<(E + 3) / 4, 256, 0, stream>>>(
        src, dst, ew, dinv, hbuf, abuf, E);
    bias_relu_kernel<D_HID><<<((size_t)N * D_HID + 255) / 256, 256, 0, stream>>>(
        abuf, b1, N * D_HID);

    // ---- layer 2: h2 = hrelu @ W2 ; agg2 = scatter(norm * h2[src])
    gemm_wmma_f32_kernel<D_IN, D_HID><<<gemm_grid, 128, 0, stream>>>(abuf, wp2, hbuf, N);
    agg_init_self_kernel<D_IN><<<((size_t)N * D_IN + 255) / 256, 256, 0, stream>>>(
        hbuf, dinv, abuf, N * D_IN);
    edge_scatter_kernel<D_IN><<<(E + 7) / 8, 256, 0, stream>>>(
        src, dst, ew, dinv, hbuf, abuf, E);

    // ---- layernorm(agg2 + b2) -> out
    layernorm_kernel<<<(N + 7) / 8, 256, 0, stream>>>(
        abuf, b2, gamma, beta, (float*)d_out, N);
}